// GCNSummarizer_2241972928674
// MI455X (gfx1250) — compile-verified
//
#include <hip/hip_runtime.h>
#include <hip/hip_bf16.h>

typedef float v2f __attribute__((ext_vector_type(2)));
typedef float v8f __attribute__((ext_vector_type(8)));

#define HID 128
#define NGRAPH 512

// ---------------- utility: zero a float region ----------------
__global__ void gcn_zero_kernel(float* __restrict__ p, long n) {
    long i = (long)blockIdx.x * blockDim.x + threadIdx.x;
    long stride = (long)gridDim.x * blockDim.x;
    for (; i < n; i += stride) p[i] = 0.0f;
}

// ---------------- degree: deg[dst] += ew ----------------
__global__ void gcn_deg_kernel(const int* __restrict__ dst,
                               const float* __restrict__ ew,
                               float* __restrict__ deg, int E) {
    int i = blockIdx.x * blockDim.x + threadIdx.x;
    int stride = gridDim.x * blockDim.x;
    for (; i < E; i += stride) atomicAdd(&deg[dst[i]], ew[i]);
}

// ---------------- dinv[i] = rsqrt(deg[i] + 1)  (self-loop weight 1; always > 0) ----
__global__ void gcn_dinv_kernel(float* __restrict__ deg_dinv, int n) {
    int i = blockIdx.x * blockDim.x + threadIdx.x;
    int stride = gridDim.x * blockDim.x;
    for (; i < n; i += stride) deg_dinv[i] = rsqrtf(deg_dinv[i] + 1.0f);
}

// ---------------- norm[e] = dinv[src]*ew*dinv[dst] ----------------
__global__ void gcn_norm_kernel(const int* __restrict__ src,
                                const int* __restrict__ dst,
                                const float* __restrict__ ew,
                                const float* __restrict__ dinv,
                                float* __restrict__ norm, int E) {
    int i = blockIdx.x * blockDim.x + threadIdx.x;
    int stride = gridDim.x * blockDim.x;
    for (; i < E; i += stride) norm[i] = dinv[src[i]] * ew[i] * dinv[dst[i]];
}

// ---------------- H = (relu?)X @ W   via V_WMMA_F32_16X16X4_F32 ----------------
// Persistent blocks: 256 threads = 8 wave32. Wave w holds its 16-column B slice
// of W register-resident (v2f bfrag[32] = 64 VGPRs) and grid-strides over 16-row
// tiles. A tile (8 KB, contiguous) is staged global->LDS with
// GLOBAL_LOAD_ASYNC_TO_LDS_B128 (ASYNCcnt) -- no VGPR round trip. RELU is a
// compile-time template parameter so the hot loop has no per-element selects.
template <int RELU>
__global__ __launch_bounds__(256) void gcn_gemm_wmma_kernel(
    const float* __restrict__ X, const float* __restrict__ W,
    float* __restrict__ H, int n) {
    __shared__ float ldsA[16 * HID];

    const int wave  = threadIdx.x >> 5;
    const int lane  = threadIdx.x & 31;
    const int n0    = wave * 16;
    const int l15   = lane & 15;          // M index (A) / N index (B)
    const int khalf = (lane >> 4) * 2;    // lanes 16-31 hold K+2,K+3

    // Preload B fragments for this wave's 16-column slice (once per block).
    // B 4x16 (KxN): lane = N column, lane-half selects the K pair.
    v2f bfrag[32];
    #pragma unroll
    for (int kk = 0; kk < 32; ++kk) {
        int k0 = kk * 4;
        bfrag[kk].x = W[(k0 + khalf) * HID + n0 + l15];
        bfrag[kk].y = W[(k0 + khalf + 1) * HID + n0 + l15];
    }

    const uint32_t lds_base = (uint32_t)(uintptr_t)ldsA;  // LDS offset = addr[31:0]
    const long     nbytes   = (long)n * HID * (long)sizeof(float);
    const int      ntiles   = (n + 15) >> 4;

    for (int t = blockIdx.x; t < ntiles; t += gridDim.x) {
        const int  m0        = t * 16;
        const long tile_byte = (long)m0 * HID * sizeof(float);

        // ---- async stage: 8 KB contiguous tile, 256 threads x 2 x 16B ----
        #pragma unroll
        for (int p = 0; p < 2; ++p) {
            uint32_t co = threadIdx.x * 16u + p * 4096u;
            long     gb = tile_byte + co;
            if (gb + 16 <= nbytes) {
                uint32_t ldsa = lds_base + co;
                uint32_t voff = (uint32_t)gb;           // n*HID*4 = 51 MB < 4 GB
                asm volatile("global_load_async_to_lds_b128 %0, %1, %2"
                             :: "v"(ldsa), "v"(voff), "s"(X) : "memory");
            } else {                                    // tail rows -> zero fill
                float4 z = {0.f, 0.f, 0.f, 0.f};
                *(float4*)((char*)ldsA + co) = z;
            }
        }
        asm volatile("s_wait_asynccnt 0x0" ::: "memory");
        __syncthreads();

        // ---- compute: 32 x v_wmma_f32_16x16x4_f32, A from LDS (+fused ReLU) ----
        v8f c = {0.f, 0.f, 0.f, 0.f, 0.f, 0.f, 0.f, 0.f};
        #pragma unroll
        for (int kk = 0; kk < 32; ++kk) {
            int k0 = kk * 4;
            v2f a;
            // A 16x4 (MxK): lane<16 -> {K=k0,k0+1}, lane>=16 -> {K=k0+2,k0+3}
            a.x = ldsA[l15 * HID + k0 + khalf];
            a.y = ldsA[l15 * HID + k0 + khalf + 1];
            if (RELU) { a.x = fmaxf(a.x, 0.f); a.y = fmaxf(a.y, 0.f); }
            c = __builtin_amdgcn_wmma_f32_16x16x4_f32(
                    /*neg_a=*/false, a, /*neg_b=*/false, bfrag[kk],
                    /*c_mod=*/(short)0, c, /*reuse_a=*/false, /*reuse_b=*/false);
        }

        // C/D layout: VGPR r -> M = r (lanes 0-15) or 8+r (lanes 16-31); N = lane&15
        const int rbase = m0 + (lane >> 4) * 8;
        const int col   = n0 + l15;
        #pragma unroll
        for (int r = 0; r < 8; ++r) {
            int row = rbase + r;
            if (row < n) H[(long)row * HID + col] = c[r];
        }
        __syncthreads();   // all waves done reading ldsA before next stage
    }
}

// ---------------- agg[i][c] = h[i][c]*dinv[i]^2 + bias[c]  (self-loop + bias init) ----
__global__ void gcn_selfloop_bias_kernel(const float* __restrict__ h,
                                         const float* __restrict__ dinv,
                                         const float* __restrict__ bias,
                                         float* __restrict__ agg, long total) {
    long i = (long)blockIdx.x * blockDim.x + threadIdx.x;
    long stride = (long)gridDim.x * blockDim.x;
    for (; i < total; i += stride) {
        int node = (int)(i >> 7);
        int c    = (int)(i & 127);
        float d  = dinv[node];
        agg[i] = h[i] * d * d + bias[c];
    }
}

// ---------------- edge scatter: agg[dst] += norm[e] * h[src], one wave per edge ----
__global__ __launch_bounds__(256) void gcn_scatter_kernel(
    const int* __restrict__ src, const int* __restrict__ dst,
    const float* __restrict__ norm, const float* __restrict__ h,
    float* __restrict__ agg, int E) {
    const int lane = threadIdx.x & 31;
    const int wid  = (blockIdx.x * blockDim.x + threadIdx.x) >> 5;
    const int nw   = (gridDim.x * blockDim.x) >> 5;
    for (int e = wid; e < E; e += nw) {
        int s = src[e], d = dst[e];
        float w = norm[e];
        if (e + nw < E)  // prefetch next source row (global_prefetch_b8)
            __builtin_prefetch(h + (long)src[e + nw] * HID + lane * 4, 0, 1);
        const float4 v = ((const float4*)(h + (long)s * HID))[lane];
        float* ad = agg + (long)d * HID + lane * 4;
        atomicAdd(ad + 0, v.x * w);
        atomicAdd(ad + 1, v.y * w);
        atomicAdd(ad + 2, v.z * w);
        atomicAdd(ad + 3, v.w * w);
    }
}

// ---------------- pooling accumulate: sums[batch[i]] += relu(agg[i]); cnt++ ----------
__global__ __launch_bounds__(256) void gcn_pool_kernel(
    const float* __restrict__ agg, const int* __restrict__ batch,
    float* __restrict__ sums, float* __restrict__ cnt, int n) {
    const int lane = threadIdx.x & 31;
    const int wid  = (blockIdx.x * blockDim.x + threadIdx.x) >> 5;
    const int nw   = (gridDim.x * blockDim.x) >> 5;
    for (int node = wid; node < n; node += nw) {
        int g = batch[node];
        float4 v = ((const float4*)(agg + (long)node * HID))[lane];
        v.x = fmaxf(v.x, 0.f); v.y = fmaxf(v.y, 0.f);
        v.z = fmaxf(v.z, 0.f); v.w = fmaxf(v.w, 0.f);
        float* sg = sums + (long)g * HID + lane * 4;
        atomicAdd(sg + 0, v.x);
        atomicAdd(sg + 1, v.y);
        atomicAdd(sg + 2, v.z);
        atomicAdd(sg + 3, v.w);
        if (lane == 0) atomicAdd(&cnt[g], 1.0f);
    }
}

// ---------------- head: out[g] = sigmoid((sums[g]/max(cnt,1)) . Wc + bc) ------------
__global__ __launch_bounds__(256) void gcn_head_kernel(
    const float* __restrict__ sums, const float* __restrict__ cnt,
    const float* __restrict__ Wc, const float* __restrict__ bc,
    float* __restrict__ out) {
    const int lane = threadIdx.x & 31;
    const int g    = (blockIdx.x * blockDim.x + threadIdx.x) >> 5;
    if (g >= NGRAPH) return;
    float inv = 1.0f / fmaxf(cnt[g], 1.0f);
    float acc = 0.0f;
    #pragma unroll
    for (int j = 0; j < 4; ++j) {
        int idx = lane * 4 + j;
        acc += sums[(long)g * HID + idx] * inv * Wc[idx];
    }
    #pragma unroll
    for (int off = 16; off > 0; off >>= 1)
        acc += __shfl_down(acc, off, 32);
    if (lane == 0) out[g] = 1.0f / (1.0f + __expf(-(acc + bc[0])));
}

extern "C" void kernel_launch(void* const* d_in, const int* in_sizes, int n_in,
                              void* d_out, int out_size, void* d_ws, size_t ws_size,
                              hipStream_t stream) {
    const float* x    = (const float*)d_in[0];
    const int*   ei   = (const int*)d_in[1];
    const float* ew   = (const float*)d_in[2];
    const int*   bat  = (const int*)d_in[3];
    const float* Wl[4] = {(const float*)d_in[4], (const float*)d_in[6],
                          (const float*)d_in[8], (const float*)d_in[10]};
    const float* bl[4] = {(const float*)d_in[5], (const float*)d_in[7],
                          (const float*)d_in[9], (const float*)d_in[11]};
    const float* Wc = (const float*)d_in[12];
    const float* bc = (const float*)d_in[13];
    float* out = (float*)d_out;

    const int n = in_sizes[0] / HID;       // 100000
    const int E = in_sizes[2];             // 3200000
    const int* src = ei;
    const int* dst = ei + E;

    // workspace carve-up (float-aligned; dinv/sums/cnt contiguous for one zero pass)
    char* w = (char*)d_ws;
    float* hbuf   = (float*)w;  w += (size_t)n * HID * sizeof(float);
    float* aggbuf = (float*)w;  w += (size_t)n * HID * sizeof(float);
    float* normbf = (float*)w;  w += (size_t)E * sizeof(float);
    float* dinv   = (float*)w;  w += (size_t)n * sizeof(float);        // also deg
    float* sums   = (float*)w;  w += (size_t)NGRAPH * HID * sizeof(float);
    float* cnt    = (float*)w;  w += (size_t)NGRAPH * sizeof(float);

    const long nzero = (long)n + (long)NGRAPH * HID + NGRAPH;
    gcn_zero_kernel<<<2048, 256, 0, stream>>>(dinv, nzero);
    gcn_deg_kernel<<<2048, 256, 0, stream>>>(dst, ew, dinv, E);
    gcn_dinv_kernel<<<(n + 255) / 256, 256, 0, stream>>>(dinv, n);
    gcn_norm_kernel<<<2048, 256, 0, stream>>>(src, dst, ew, dinv, normbf, E);

    const int gemm_blocks    = 640;                  // persistent, grid-stride over tiles
    const long total         = (long)n * HID;
    const int scatter_blocks = 2048;                 // 16384 waves, grid-stride over edges

    const float* layer_in = x;
    for (int l = 0; l < 4; ++l) {
        if (l == 0)
            gcn_gemm_wmma_kernel<0><<<gemm_blocks, 256, 0, stream>>>(layer_in, Wl[l], hbuf, n);
        else
            gcn_gemm_wmma_kernel<1><<<gemm_blocks, 256, 0, stream>>>(layer_in, Wl[l], hbuf, n);
        gcn_selfloop_bias_kernel<<<4096, 256, 0, stream>>>(hbuf, dinv, bl[l], aggbuf, total);
        gcn_scatter_kernel<<<scatter_blocks, 256, 0, stream>>>(
            src, dst, normbf, hbuf, aggbuf, E);
        layer_in = aggbuf;   // ReLU fused into next consumer's read
    }

    gcn_pool_kernel<<<2048, 256, 0, stream>>>(aggbuf, bat, sums, cnt, n);
    gcn_head_kernel<<<(NGRAPH * 32 + 255) / 256, 256, 0, stream>>>(sums, cnt, Wc, bc, out);
}